// PhysicsAttentionBlock_14121852469806
// MI455X (gfx1250) — compile-verified
//
#include <hip/hip_runtime.h>
#include <hip/hip_bf16.h>

// ---------------- problem constants ----------------
static constexpr int Bz  = 4;
static constexpr int Nn  = 32768;
static constexpr int Hh  = 256;
static constexpr int Ss  = 64;
static constexpr int NH  = 8;
static constexpr int HD  = 32;

typedef float v2f __attribute__((ext_vector_type(2)));
typedef float v8f __attribute__((ext_vector_type(8)));

__device__ __forceinline__ v8f wmma4(v2f a, v2f b, v8f c) {
  // V_WMMA_F32_16X16X4_F32 : D = A(16x4) * B(4x16) + C(16x16)
  return __builtin_amdgcn_wmma_f32_16x16x4_f32(
      /*neg_a=*/false, a, /*neg_b=*/false, b,
      /*c_mod=*/(short)0, c, /*reuse_a=*/false, /*reuse_b=*/false);
}

__device__ __forceinline__ v8f v8zero() {
  v8f z = {0.f,0.f,0.f,0.f,0.f,0.f,0.f,0.f};
  return z;
}

// =====================================================================
// Kernel 1: logits -> softmax -> W (in LDS) -> partial st = W^T @ x
// grid = 128 blocks (4 batches x 32 chunks of 1024 tokens), 256 threads.
// =====================================================================
__global__ void __launch_bounds__(256, 1)
k1_pool(const float* __restrict__ x,
        const float* __restrict__ sq_w,
        const float* __restrict__ sq_b,
        float* __restrict__ stp) {
  __shared__ float wlds[128 * 72];   // 128 tokens x 64 slots, stride 72 (36.9 KB)

  const int tid  = threadIdx.x;
  const int wave = tid >> 5;
  const int l    = tid & 31;
  const int l16  = l & 15;
  const int half = l >> 4;

  const int blk   = blockIdx.x;
  const int b     = blk >> 5;        // batch
  const int chunk = blk & 31;        // 1024-token chunk within batch
  const size_t xb = (size_t)b * Nn * Hh;

  // pool ownership: wave w owns ALL 4 s-blocks x 2 h-blocks (h in [32w,32w+32))
  const int hw = 32 * wave;

  v8f cc[8];                         // [s_blk*2 + hb]
  #pragma unroll
  for (int j = 0; j < 8; ++j) cc[j] = v8zero();

  float sb[4];
  #pragma unroll
  for (int t = 0; t < 4; ++t) sb[t] = sq_b[16 * t + l16];

  const float* wr0 = sq_w + (size_t)l16 * Hh + 2 * half;   // slot-row base (t stride 16*Hh)

  for (int tile = 0; tile < 8; ++tile) {           // 8 x 128-token tiles
    const int tok0  = chunk * 1024 + tile * 128;
    const int myTok = tok0 + 16 * wave;            // this wave's 16 tokens

    // ---- phase A: logits (16 tokens x 64 slots), double-buffered ----
    v8f c4[4];
    #pragma unroll
    for (int t = 0; t < 4; ++t) c4[t] = v8zero();

    const float* ar = x + xb + (size_t)(myTok + l16) * Hh + 2 * half;

    v2f an, bn0, bn1, bn2, bn3;
    auto ldA = [&](int k0) {
      an  = *(const v2f*)(ar + k0);
      bn0 = *(const v2f*)(wr0 + k0);
      bn1 = *(const v2f*)(wr0 + 16 * Hh + k0);
      bn2 = *(const v2f*)(wr0 + 32 * Hh + k0);
      bn3 = *(const v2f*)(wr0 + 48 * Hh + k0);
    };
    ldA(0);
    for (int kt = 0; kt < 64; ++kt) {
      v2f ac = an, b0 = bn0, b1 = bn1, b2 = bn2, b3 = bn3;
      const int kn = (kt < 63) ? (kt + 1) * 4 : 252;   // clamped prefetch
      ldA(kn);
      c4[0] = wmma4(ac, b0, c4[0]);
      c4[1] = wmma4(ac, b1, c4[1]);
      c4[2] = wmma4(ac, b2, c4[2]);
      c4[3] = wmma4(ac, b3, c4[3]);
    }

    // ---- softmax over 64 slots per token row ----
    #pragma unroll
    for (int r = 0; r < 8; ++r) {
      float m = -3.0e38f;
      #pragma unroll
      for (int t = 0; t < 4; ++t) { c4[t][r] += sb[t]; m = fmaxf(m, c4[t][r]); }
      #pragma unroll
      for (int msk = 1; msk < 16; msk <<= 1) m = fmaxf(m, __shfl_xor(m, msk));
      float s = 0.f;
      #pragma unroll
      for (int t = 0; t < 4; ++t) { float e = __expf(c4[t][r] - m); c4[t][r] = e; s += e; }
      #pragma unroll
      for (int msk = 1; msk < 16; msk <<= 1) s += __shfl_xor(s, msk);
      const float inv = 1.f / s;
      #pragma unroll
      for (int t = 0; t < 4; ++t) c4[t][r] *= inv;
    }

    __syncthreads();   // previous phase-B readers done before overwrite
    #pragma unroll
    for (int r = 0; r < 8; ++r) {
      const int tokL = 16 * wave + r + 8 * half;
      #pragma unroll
      for (int t = 0; t < 4; ++t)
        wlds[tokL * 72 + 16 * t + l16] = c4[t][r];
    }
    __syncthreads();

    // ---- phase B: cc += W^T(64 x 128) @ x(128 x 32-strip), double-buffered ----
    v2f aan[4], bbn[2];
    auto ldB = [&](int k0) {
      #pragma unroll
      for (int sbk = 0; sbk < 4; ++sbk) {
        aan[sbk].x = wlds[(k0     + 2 * half) * 72 + 16 * sbk + l16];
        aan[sbk].y = wlds[(k0 + 1 + 2 * half) * 72 + 16 * sbk + l16];
      }
      const float* p0 = x + xb + (size_t)(tok0 + k0 + 2 * half) * Hh + hw + l16;
      #pragma unroll
      for (int hb = 0; hb < 2; ++hb) {
        bbn[hb].x = p0[16 * hb];
        bbn[hb].y = p0[Hh + 16 * hb];
      }
    };
    ldB(0);
    for (int kt = 0; kt < 32; ++kt) {
      v2f aa[4], bbv[2];
      #pragma unroll
      for (int sbk = 0; sbk < 4; ++sbk) aa[sbk] = aan[sbk];
      #pragma unroll
      for (int hb = 0; hb < 2; ++hb) bbv[hb] = bbn[hb];
      const int kn = (kt < 31) ? (kt + 1) * 4 : 124;   // clamped prefetch
      ldB(kn);
      #pragma unroll
      for (int sbk = 0; sbk < 4; ++sbk)
        #pragma unroll
        for (int hb = 0; hb < 2; ++hb)
          cc[2 * sbk + hb] = wmma4(aa[sbk], bbv[hb], cc[2 * sbk + hb]);
    }
  }

  // ---- write partial st for this block ----
  #pragma unroll
  for (int sbk = 0; sbk < 4; ++sbk) {
    #pragma unroll
    for (int hb = 0; hb < 2; ++hb) {
      #pragma unroll
      for (int r = 0; r < 8; ++r) {
        const int s = 16 * sbk + r + 8 * half;
        const int h = hw + 16 * hb + l16;
        stp[((size_t)blk * Ss + s) * Hh + h] = cc[2 * sbk + hb][r];
      }
    }
  }
}

// =====================================================================
// Kernel 2: reduce 32 chunk-partials per batch -> st0 [B,S,H]
// =====================================================================
__global__ void k2_reduce(const float* __restrict__ stp, float* __restrict__ st0) {
  const int i = blockIdx.x * 256 + threadIdx.x;   // 65536 elements
  const int b = i >> 14;
  const int j = i & 16383;
  float s = 0.f;
  for (int c = 0; c < 32; ++c)
    s += stp[((size_t)(b * 32 + c) << 14) + j];
  st0[i] = s;
}

// =====================================================================
// Generic one-wave WMMA GEMM on slice tokens:
//   out[row, col] = epilogue( A[row,:] . Wt[col,:] + bias[col] )
// grid = (rows/16, outs/64), block = 32 (one wave).
// mode 0: (+resid) ; mode 1: exact GELU ; mode 2: resid + sigmoid(.)*es
// =====================================================================
__global__ void __launch_bounds__(32, 1)
slice_gemm(const float* __restrict__ A,
           const float* __restrict__ Wt,
           const float* __restrict__ bias,
           const float* __restrict__ resid,
           const float* __restrict__ es,
           float* __restrict__ out,
           int K, int outs, int mode) {
  const int l    = threadIdx.x & 31;
  const int l16  = l & 15;
  const int half = l >> 4;
  const int row0 = blockIdx.x * 16;
  const int col0 = blockIdx.y * 64;

  v8f c[4];
  #pragma unroll
  for (int t = 0; t < 4; ++t) c[t] = v8zero();

  const float* arow = A + (size_t)(row0 + l16) * K + 2 * half;
  const float* wrow = Wt + (size_t)(col0 + l16) * K + 2 * half;   // t stride 16*K

  v2f an, bn[4];
  auto ld = [&](int k0) {
    an = *(const v2f*)(arow + k0);
    #pragma unroll
    for (int t = 0; t < 4; ++t)
      bn[t] = *(const v2f*)(wrow + (size_t)(16 * t) * K + k0);
  };
  ld(0);
  for (int k0 = 0; k0 < K; k0 += 4) {
    v2f ac = an, b0 = bn[0], b1 = bn[1], b2 = bn[2], b3 = bn[3];
    const int kn = (k0 + 4 < K) ? k0 + 4 : k0;   // clamped prefetch
    ld(kn);
    c[0] = wmma4(ac, b0, c[0]);
    c[1] = wmma4(ac, b1, c[1]);
    c[2] = wmma4(ac, b2, c[2]);
    c[3] = wmma4(ac, b3, c[3]);
  }

  #pragma unroll
  for (int t = 0; t < 4; ++t) {
    const int col = col0 + 16 * t + l16;
    #pragma unroll
    for (int r = 0; r < 8; ++r) {
      const int row = row0 + r + 8 * half;
      float v = c[t][r] + bias[col];
      if (mode == 1) {
        v = 0.5f * v * (1.f + erff(v * 0.70710678118654752f));
      } else if (mode == 2) {
        const float g = 1.f / (1.f + __expf(-v));
        v = resid[(size_t)row * outs + col] + g * es[(size_t)(row & (Ss - 1)) * outs + col];
      } else if (resid) {
        v += resid[(size_t)row * outs + col];
      }
      out[(size_t)row * outs + col] = v;
    }
  }
}

// =====================================================================
// Attention over slice tokens: grid = B*NH blocks, 64 threads.
// =====================================================================
__global__ void k_attn(const float* __restrict__ qkv, float* __restrict__ ao) {
  __shared__ float q[Ss][HD + 1], kk[Ss][HD + 1], vv[Ss][HD + 1];
  __shared__ float sc[Ss][Ss + 1];

  const int blk  = blockIdx.x;
  const int b    = blk >> 3;
  const int head = blk & 7;
  const int t    = threadIdx.x;          // query index 0..63

  const float* base = qkv + (size_t)(b * Ss + t) * (3 * Hh) + head * HD;
  for (int d = 0; d < HD; ++d) {
    q[t][d]  = base[d];
    kk[t][d] = base[Hh + d];
    vv[t][d] = base[2 * Hh + d];
  }
  __syncthreads();

  float m = -3.0e38f;
  for (int j = 0; j < Ss; ++j) {
    float s = 0.f;
    #pragma unroll
    for (int d = 0; d < HD; ++d) s += q[t][d] * kk[j][d];
    s *= 0.17677669529663687f;           // 1/sqrt(32)
    sc[t][j] = s;
    m = fmaxf(m, s);
  }
  float sum = 0.f;
  for (int j = 0; j < Ss; ++j) { float e = __expf(sc[t][j] - m); sc[t][j] = e; sum += e; }
  const float inv = 1.f / sum;

  float* o = ao + (size_t)(b * Ss + t) * Hh + head * HD;
  for (int d = 0; d < HD; ++d) {
    float s = 0.f;
    for (int j = 0; j < Ss; ++j) s += sc[t][j] * vv[j][d];
    o[d] = s * inv;
  }
}

// =====================================================================
// LayerNorm over H=256, one thread per row (256 rows).
// =====================================================================
__global__ void k_ln(const float* __restrict__ y,
                     const float* __restrict__ g,
                     const float* __restrict__ bb,
                     float* __restrict__ out) {
  const int row = threadIdx.x;
  const float* p = y + (size_t)row * Hh;
  float mu = 0.f;
  for (int j = 0; j < Hh; ++j) mu += p[j];
  mu *= (1.f / Hh);
  float var = 0.f;
  for (int j = 0; j < Hh; ++j) { float d = p[j] - mu; var += d * d; }
  var *= (1.f / Hh);
  const float inv = rsqrtf(var + 1e-5f);
  float* o = out + (size_t)row * Hh;
  for (int j = 0; j < Hh; ++j) o[j] = (p[j] - mu) * inv * g[j] + bb[j];
}

// =====================================================================
// Kernel 4: decode. Recompute W per 16-token tile, out = W @ st2 + x.
// grid = 1024 blocks (4 x 256 tiles of 128 tokens), 256 threads.
// =====================================================================
__global__ void __launch_bounds__(256, 1)
k4_decode(const float* __restrict__ x,
          const float* __restrict__ sq_w,
          const float* __restrict__ sq_b,
          const float* __restrict__ st2,
          float* __restrict__ out) {
  __shared__ float wlds[8 * 16 * 72];

  const int tid  = threadIdx.x;
  const int wave = tid >> 5;
  const int l    = tid & 31;
  const int l16  = l & 15;
  const int half = l >> 4;

  const int blk  = blockIdx.x;
  const int b    = blk >> 8;
  const int tok0 = (blk & 255) * 128;
  const size_t xb = (size_t)b * Nn * Hh;
  const int myTok = tok0 + 16 * wave;

  // ---- logits (double-buffered) ----
  v8f c4[4];
  #pragma unroll
  for (int t = 0; t < 4; ++t) c4[t] = v8zero();
  float sb[4];
  #pragma unroll
  for (int t = 0; t < 4; ++t) sb[t] = sq_b[16 * t + l16];

  const float* ar  = x + xb + (size_t)(myTok + l16) * Hh + 2 * half;
  const float* wr0 = sq_w + (size_t)l16 * Hh + 2 * half;

  {
    v2f an, bn0, bn1, bn2, bn3;
    auto ldA = [&](int k0) {
      an  = *(const v2f*)(ar + k0);
      bn0 = *(const v2f*)(wr0 + k0);
      bn1 = *(const v2f*)(wr0 + 16 * Hh + k0);
      bn2 = *(const v2f*)(wr0 + 32 * Hh + k0);
      bn3 = *(const v2f*)(wr0 + 48 * Hh + k0);
    };
    ldA(0);
    for (int kt = 0; kt < 64; ++kt) {
      v2f ac = an, b0 = bn0, b1 = bn1, b2 = bn2, b3 = bn3;
      const int kn = (kt < 63) ? (kt + 1) * 4 : 252;
      ldA(kn);
      c4[0] = wmma4(ac, b0, c4[0]);
      c4[1] = wmma4(ac, b1, c4[1]);
      c4[2] = wmma4(ac, b2, c4[2]);
      c4[3] = wmma4(ac, b3, c4[3]);
    }
  }

  // ---- softmax ----
  #pragma unroll
  for (int r = 0; r < 8; ++r) {
    float m = -3.0e38f;
    #pragma unroll
    for (int t = 0; t < 4; ++t) { c4[t][r] += sb[t]; m = fmaxf(m, c4[t][r]); }
    #pragma unroll
    for (int msk = 1; msk < 16; msk <<= 1) m = fmaxf(m, __shfl_xor(m, msk));
    float s = 0.f;
    #pragma unroll
    for (int t = 0; t < 4; ++t) { float e = __expf(c4[t][r] - m); c4[t][r] = e; s += e; }
    #pragma unroll
    for (int msk = 1; msk < 16; msk <<= 1) s += __shfl_xor(s, msk);
    const float inv = 1.f / s;
    #pragma unroll
    for (int t = 0; t < 4; ++t) c4[t][r] *= inv;
  }

  // ---- stage W (per-wave region) so it can be re-read in A-operand layout ----
  #pragma unroll
  for (int r = 0; r < 8; ++r) {
    const int tokL = 16 * wave + r + 8 * half;
    #pragma unroll
    for (int t = 0; t < 4; ++t)
      wlds[tokL * 72 + 16 * t + l16] = c4[t][r];
  }
  __syncthreads();

  // ---- out = W(16x64) @ st2(64x256) + x, two 128-col halves ----
  const float* s2 = st2 + (size_t)b * Ss * Hh;
  for (int hh = 0; hh < 2; ++hh) {
    const int hbase = hh * 128;
    v8f cc[8];
    #pragma unroll
    for (int j = 0; j < 8; ++j) cc[j] = v8zero();

    v2f an; v2f bn[8];
    auto ldO = [&](int k0) {
      an = *(const v2f*)(&wlds[(wave * 16 + l16) * 72 + k0 + 2 * half]);
      const float* p0 = s2 + (size_t)(k0 + 2 * half) * Hh + hbase + l16;
      #pragma unroll
      for (int j = 0; j < 8; ++j) {
        bn[j].x = p0[16 * j];
        bn[j].y = p0[Hh + 16 * j];
      }
    };
    ldO(0);
    for (int kt = 0; kt < 16; ++kt) {
      v2f ac = an; v2f bv[8];
      #pragma unroll
      for (int j = 0; j < 8; ++j) bv[j] = bn[j];
      const int kn = (kt < 15) ? (kt + 1) * 4 : 60;
      ldO(kn);
      #pragma unroll
      for (int j = 0; j < 8; ++j) cc[j] = wmma4(ac, bv[j], cc[j]);
    }

    #pragma unroll
    for (int j = 0; j < 8; ++j) {
      #pragma unroll
      for (int r = 0; r < 8; ++r) {
        const int tokL = 16 * wave + r + 8 * half;
        const int h = hbase + 16 * j + l16;
        const size_t idx = xb + (size_t)(tok0 + tokL) * Hh + h;
        out[idx] = cc[j][r] + x[idx];
      }
    }
  }
}

// =====================================================================
// Host-side launch
// =====================================================================
extern "C" void kernel_launch(void* const* d_in, const int* in_sizes, int n_in,
                              void* d_out, int out_size, void* d_ws, size_t ws_size,
                              hipStream_t stream) {
  const float* x        = (const float*)d_in[0];
  const float* sq_w     = (const float*)d_in[1];
  const float* sq_b     = (const float*)d_in[2];
  const float* eid_st   = (const float*)d_in[3];
  const float* eid_gw   = (const float*)d_in[4];
  const float* eid_gb   = (const float*)d_in[5];
  const float* in_w     = (const float*)d_in[6];
  const float* in_b     = (const float*)d_in[7];
  const float* out_w    = (const float*)d_in[8];
  const float* out_b    = (const float*)d_in[9];
  const float* n1_g     = (const float*)d_in[10];
  const float* n1_b     = (const float*)d_in[11];
  const float* n2_g     = (const float*)d_in[12];
  const float* n2_b     = (const float*)d_in[13];
  const float* f1_w     = (const float*)d_in[14];
  const float* f1_b     = (const float*)d_in[15];
  const float* f2_w     = (const float*)d_in[16];
  const float* f2_b     = (const float*)d_in[17];
  float* out = (float*)d_out;

  // workspace carve-up (floats)
  float* w   = (float*)d_ws;
  float* stp = w;                         // 128 * 64 * 256  = 2,097,152
  float* st0 = stp + 2097152;             // 65536
  float* stg = st0 + 65536;               // 65536
  float* qkv = stg + 65536;               // 196608
  float* ao  = qkv + 196608;              // 65536
  float* y1  = ao  + 65536;               // 65536
  float* st1 = y1  + 65536;               // 65536
  float* h1  = st1 + 65536;               // 262144
  float* y2  = h1  + 262144;              // 65536
  float* st2 = y2  + 65536;               // 65536

  // 1) softmax-pool (split-K partials), 2) deterministic reduce
  k1_pool<<<dim3(128), dim3(256), 0, stream>>>(x, sq_w, sq_b, stp);
  k2_reduce<<<dim3(256), dim3(256), 0, stream>>>(stp, st0);

  // 3) eidetic gating: stg = st0 + sigmoid(st0@gw^T + gb) * eid_state
  slice_gemm<<<dim3(16, 4), dim3(32), 0, stream>>>(st0, eid_gw, eid_gb, st0, eid_st,
                                                   stg, Hh, Hh, 2);
  // 4) qkv
  slice_gemm<<<dim3(16, 12), dim3(32), 0, stream>>>(stg, in_w, in_b, nullptr, nullptr,
                                                    qkv, Hh, 3 * Hh, 0);
  // 5) attention
  k_attn<<<dim3(Bz * NH), dim3(64), 0, stream>>>(qkv, ao);
  // 6) out proj + residual
  slice_gemm<<<dim3(16, 4), dim3(32), 0, stream>>>(ao, out_w, out_b, stg, nullptr,
                                                   y1, Hh, Hh, 0);
  // 7) LN1
  k_ln<<<dim3(1), dim3(256), 0, stream>>>(y1, n1_g, n1_b, st1);
  // 8) FFN up + exact GELU
  slice_gemm<<<dim3(16, 16), dim3(32), 0, stream>>>(st1, f1_w, f1_b, nullptr, nullptr,
                                                    h1, Hh, 4 * Hh, 1);
  // 9) FFN down + residual
  slice_gemm<<<dim3(16, 4), dim3(32), 0, stream>>>(h1, f2_w, f2_b, st1, nullptr,
                                                   y2, 4 * Hh, Hh, 0);
  // 10) LN2
  k_ln<<<dim3(1), dim3(256), 0, stream>>>(y2, n2_g, n2_b, st2);
  // 11) decode (recompute W) + residual x
  k4_decode<<<dim3(1024), dim3(256), 0, stream>>>(x, sq_w, sq_b, st2, out);
}